// PLRNN_88124138979737
// MI455X (gfx1250) — compile-verified
//
#include <hip/hip_runtime.h>

#define T_DIM 2048
#define B_DIM 64
#define IN_DIM 256
#define H_DIM 512
#define SCAN_BLOCKS 16
#define SCAN_THREADS 256

typedef __attribute__((ext_vector_type(2))) float v2f;
typedef __attribute__((ext_vector_type(8))) float v8f;

#define WMMA_F32(A, B, C) \
    __builtin_amdgcn_wmma_f32_16x16x4_f32(false, (A), false, (B), (short)0, (C), false, false)

// ---------------------------------------------------------------------------
// Prep: pair-packed, K-major operand layouts so every WMMA B-fetch is one
// 8-byte load of an even/odd K pair:
//   Wp [kp][n][2] : Wp[kp][n][j]  = W[n][2kp+j] * (n != 2kp+j)   (H/2 x H x 2)
//   Cwp[kp][n][2] : Cwp[kp][n][j] = C_w[n][2kp+j]               (IN/2 x H x 2)
// ---------------------------------------------------------------------------
__global__ void prep_kernel(const float* __restrict__ W,
                            const float* __restrict__ C_w,
                            float* __restrict__ Wp,
                            float* __restrict__ Cwp) {
    int idx = blockIdx.x * blockDim.x + threadIdx.x;
    if (idx < (H_DIM / 2) * H_DIM) {
        int kp = idx / H_DIM, n = idx % H_DIM;
        int k0 = 2 * kp, k1 = 2 * kp + 1;
        float w0 = (n == k0) ? 0.0f : W[n * H_DIM + k0];
        float w1 = (n == k1) ? 0.0f : W[n * H_DIM + k1];
        ((float2*)Wp)[idx] = make_float2(w0, w1);
    }
    if (idx < (IN_DIM / 2) * H_DIM) {
        int kp = idx / H_DIM, n = idx % H_DIM;
        ((float2*)Cwp)[idx] = make_float2(C_w[n * IN_DIM + 2 * kp],
                                          C_w[n * IN_DIM + 2 * kp + 1]);
    }
}

// ---------------------------------------------------------------------------
// Phase 1: Cx = input @ C_w.T + C_b written straight into d_out[t,b,h].
// One wave per 16x16 tile; A-operand float2 (b64) loads, B-operand pair-packed
// b64 loads. Fully parallel: 262144 waves.
// ---------------------------------------------------------------------------
__global__ void cx_kernel(const float* __restrict__ in,
                          const float* __restrict__ Cwp,
                          const float* __restrict__ C_b,
                          float* __restrict__ out) {
    const int lane = threadIdx.x & 31;
    const int wave = threadIdx.x >> 5;
    const int wid  = blockIdx.x * (blockDim.x >> 5) + wave;
    const int NT   = H_DIM / 16;
    const int mt   = wid / NT;
    const int nt   = wid % NT;
    const int mbase = mt * 16;
    const int nbase = nt * 16;
    const int lo = lane & 15, hi = lane >> 4;

    const int arow = mbase + lo;              // A row   (M = lane%16)
    const int ncol = nbase + lo;              // B/C col (N = lane%16)

    v8f acc0 = {}, acc1 = {};
    const float* aptr = in  + (size_t)arow * IN_DIM + 2 * hi;
    const float* bptr = Cwp + ((size_t)hi * H_DIM + ncol) * 2;  // kp = k/2 + hi

    #pragma unroll 4
    for (int k = 0; k < IN_DIM; k += 8) {
        v2f a0 = *(const v2f*)(aptr + k);
        v2f b0 = *(const v2f*)(bptr + (size_t)(k >> 1) * (H_DIM * 2));
        acc0 = WMMA_F32(a0, b0, acc0);
        v2f a1 = *(const v2f*)(aptr + k + 4);
        v2f b1 = *(const v2f*)(bptr + (size_t)((k >> 1) + 2) * (H_DIM * 2));
        acc1 = WMMA_F32(a1, b1, acc1);
    }

    const float bias = C_b[ncol];
    #pragma unroll
    for (int v = 0; v < 8; ++v) {
        int r = mbase + v + 8 * hi;           // C/D row: M = v + 8*(lane/16)
        out[(size_t)r * H_DIM + ncol] = acc0[v] + acc1[v] + bias;
    }
}

// ---------------------------------------------------------------------------
// Phase 2: sequential scan. 16 blocks x 256 threads = 128 waves, one wave per
// 16x16 tile of the (64 x 512) state.
//   LDS: 256 KB pair-packed W column window + 32 KB relu(h_prev) batch tile.
//   All WMMA operands come from LDS as single ds_load_b64 per operand.
//   Device-wide step barrier: monotonic atomic counter (16 resident blocks).
// ---------------------------------------------------------------------------
__global__ void scan_kernel(const float* __restrict__ hx,
                            const float* __restrict__ Avec,
                            const float* __restrict__ Wp,
                            float* __restrict__ out,
                            int* __restrict__ counter) {
    extern __shared__ float smem[];
    float* ldsW = smem;                            // [256 kp][128 n][2] = 65536 f
    float* ldsH = smem + (H_DIM / 2) * 128 * 2;    // [16 m][512 k]     =  8192 f

    const int lane = threadIdx.x & 31;
    const int wave = threadIdx.x >> 5;
    const int w    = blockIdx.x * (SCAN_THREADS / 32) + wave;   // 0..127
    const int mt   = w >> 5;                   // 0..3  (batch tile; same per block)
    const int nt   = w & 31;                   // 0..31 (hidden tile)
    const int mbase = mt * 16;
    const int nbase = nt * 16;
    const int lo = lane & 15, hi = lane >> 4;
    const int colbase = (blockIdx.x & 3) * 128;    // block's W column window

    // Stage pair-packed W window: 256 kp-rows x 256 floats, coalesced float4.
    for (int i = threadIdx.x; i < (H_DIM / 2) * 64; i += blockDim.x) {
        int kp = i >> 6;                       // 64 float4 per kp-row
        int c  = i & 63;
        ((float4*)ldsW)[kp * 64 + c] =
            ((const float4*)Wp)[((size_t)kp * H_DIM + colbase) / 2 + c];
    }

    const int   ncol  = nbase + lo;
    const float decay = Avec[ncol];
    const int   nl2   = (wave * 16 + lo) * 2;      // local B column offset (*2)

    for (int t = 0; t < T_DIM; ++t) {
        const float* hprev = (t == 0) ? hx
                                      : out + (size_t)(t - 1) * B_DIM * H_DIM;
        float* hout = out + (size_t)t * B_DIM * H_DIM;   // holds Cx[t] on entry

        // Stage relu(h_prev) batch tile: contiguous 32 KB, relu applied once.
        {
            const float4* src = (const float4*)(hprev + (size_t)mbase * H_DIM);
            float4* dst = (float4*)ldsH;
            for (int i = threadIdx.x; i < 16 * H_DIM / 4; i += blockDim.x) {
                float4 v = src[i];
                v.x = fmaxf(v.x, 0.0f); v.y = fmaxf(v.y, 0.0f);
                v.z = fmaxf(v.z, 0.0f); v.w = fmaxf(v.w, 0.0f);
                dst[i] = v;
            }
        }
        __syncthreads();

        v8f acc0 = {}, acc1 = {}, acc2 = {}, acc3 = {};
        const float* ap = ldsH + lo * H_DIM + 2 * hi;
        const float* bp = ldsW + hi * 256 + nl2;       // kp = k/2 + hi

        #pragma unroll 4
        for (int k = 0; k < H_DIM; k += 16) {
            v2f a0 = *(const v2f*)(ap + k);
            v2f b0 = *(const v2f*)(bp + (k >> 1) * 256);
            acc0 = WMMA_F32(a0, b0, acc0);
            v2f a1 = *(const v2f*)(ap + k + 4);
            v2f b1 = *(const v2f*)(bp + ((k >> 1) + 2) * 256);
            acc1 = WMMA_F32(a1, b1, acc1);
            v2f a2 = *(const v2f*)(ap + k + 8);
            v2f b2 = *(const v2f*)(bp + ((k >> 1) + 4) * 256);
            acc2 = WMMA_F32(a2, b2, acc2);
            v2f a3 = *(const v2f*)(ap + k + 12);
            v2f b3 = *(const v2f*)(bp + ((k >> 1) + 6) * 256);
            acc3 = WMMA_F32(a3, b3, acc3);
        }

        #pragma unroll
        for (int v = 0; v < 8; ++v) {
            int r = mbase + v + 8 * hi;        // batch row
            size_t off = (size_t)r * H_DIM + ncol;
            float hp  = hprev[off];
            float val = decay * hp + ((acc0[v] + acc1[v]) + (acc2[v] + acc3[v]))
                        + hout[off];
            hout[off] = val;
            if (t == T_DIM - 1)
                out[(size_t)T_DIM * B_DIM * H_DIM + off] = val;   // h_last
        }

        // ---- device-wide step barrier (monotonic counter) ----
        __threadfence();
        __syncthreads();
        if (threadIdx.x == 0) {
            atomicAdd(counter, 1);
            const int target = SCAN_BLOCKS * (t + 1);
            while (__hip_atomic_load(counter, __ATOMIC_ACQUIRE,
                                     __HIP_MEMORY_SCOPE_AGENT) < target) {
                __builtin_amdgcn_s_sleep(1);
            }
        }
        __syncthreads();
    }
}

// ---------------------------------------------------------------------------
extern "C" void kernel_launch(void* const* d_in, const int* in_sizes, int n_in,
                              void* d_out, int out_size, void* d_ws, size_t ws_size,
                              hipStream_t stream) {
    const float* input = (const float*)d_in[0];
    const float* hx    = (const float*)d_in[1];
    const float* A     = (const float*)d_in[2];
    const float* W     = (const float*)d_in[3];
    const float* C_w   = (const float*)d_in[4];
    const float* C_b   = (const float*)d_in[5];
    float* out = (float*)d_out;

    char* ws = (char*)d_ws;
    int*   counter = (int*)ws;
    float* Wp      = (float*)(ws + 256);                                   // 1 MB
    float* Cwp     = (float*)(ws + 256 + (size_t)H_DIM * H_DIM * sizeof(float)); // 512 KB

    hipMemsetAsync(counter, 0, sizeof(int), stream);

    prep_kernel<<<((H_DIM / 2) * H_DIM + 255) / 256, 256, 0, stream>>>(W, C_w, Wp, Cwp);

    const int cx_tiles = (T_DIM * B_DIM / 16) * (H_DIM / 16);   // 262144 waves
    cx_kernel<<<cx_tiles / 8, 256, 0, stream>>>(input, Cwp, C_b, out);

    const size_t lds_bytes = ((size_t)(H_DIM / 2) * 128 * 2 + 16 * H_DIM) * sizeof(float);
    scan_kernel<<<SCAN_BLOCKS, SCAN_THREADS, lds_bytes, stream>>>(hx, A, Wp, out, counter);
}